// MultiHeadAttention_18992345383033
// MI455X (gfx1250) — compile-verified
//
#include <hip/hip_runtime.h>
#include <hip/hip_bf16.h>

typedef float v2f __attribute__((ext_vector_type(2)));
typedef float v4f __attribute__((ext_vector_type(4)));
typedef float v8f __attribute__((ext_vector_type(8)));

#define B_    2
#define S_    2048
#define D_    768
#define H_    12
#define KS_   64
#define NB_   16
#define MAXD_ 1000
#define TABW_ (2 * MAXD_ + 1)

#define LDSTRIDE 2052                         // 2048 + 4 pad: rows 8 apart -> +32 banks
#define LDS_FLOATS (16 * LDSTRIDE + 4 * 256)  // row-block + 4-tile split-K scratch

// ---------------------------------------------------------------------------
// f32 WMMA 16x16x4 (wave32). A: 16x4, B: 4x16, C/D: 16x16 f32.
// ---------------------------------------------------------------------------
__device__ __forceinline__ v8f wmma_f32(v2f a, v2f b, v8f c) {
    return __builtin_amdgcn_wmma_f32_16x16x4_f32(
        false, a, false, b, (short)0, c, false, false);
}

// ---------------------------------------------------------------------------
// Bias table: tab[h][rel] = sum_b feats[rel][b] * w[h][b]
// ---------------------------------------------------------------------------
__global__ void bias_kernel(const float* __restrict__ bw,   // [H, NB]
                            const float* __restrict__ bf,   // [TABW, NB]
                            float* __restrict__ tab) {      // [H, TABW]
    int t = blockIdx.x * blockDim.x + threadIdx.x;
    if (t >= H_ * TABW_) return;
    int h = t / TABW_;
    int r = t % TABW_;
    float s = 0.f;
#pragma unroll
    for (int i = 0; i < NB_; ++i) s += bf[r * NB_ + i] * bw[h * NB_ + i];
    tab[t] = s;
}

// ---------------------------------------------------------------------------
// Y[M,N] = X[M,K] @ W[N,K]^T. One wave computes a 16x64 strip (4 WMMA tiles),
// A fragment reused 4x. Software-pipelined: next k-step's 5 fragments are in
// flight while the current step's 4 WMMAs execute (waits retire at loadcnt<=5).
// ---------------------------------------------------------------------------
__global__ void gemm_xwT4_kernel(const float* __restrict__ X,
                                 const float* __restrict__ W,
                                 float* __restrict__ Y,
                                 int M, int N, int K) {
    int wave = (int)((blockIdx.x * blockDim.x + threadIdx.x) >> 5);
    int lane = threadIdx.x & 31;
    int n4 = N >> 6;                      // 64-wide column groups
    int mt = wave / n4, ng = wave % n4;
    int m0 = mt << 4, n0 = ng << 6;
    int half = lane >> 4, l16 = lane & 15;

    const float* arow = X + (size_t)(m0 + l16) * K + 2 * half;
    const float* brow = W + (size_t)(n0 + l16) * K + 2 * half;
    const size_t s16 = (size_t)16 * K;

    v8f c0 = {}, c1 = {}, c2 = {}, c3 = {};

    // prologue
    v2f a  = *(const v2f*)(arow);
    v2f b0 = *(const v2f*)(brow);
    v2f b1 = *(const v2f*)(brow + s16);
    v2f b2 = *(const v2f*)(brow + 2 * s16);
    v2f b3 = *(const v2f*)(brow + 3 * s16);

    for (int kk = 4; kk < K; kk += 4) {
        v2f an  = *(const v2f*)(arow + kk);
        v2f bn0 = *(const v2f*)(brow + kk);
        v2f bn1 = *(const v2f*)(brow + s16 + kk);
        v2f bn2 = *(const v2f*)(brow + 2 * s16 + kk);
        v2f bn3 = *(const v2f*)(brow + 3 * s16 + kk);
        c0 = wmma_f32(a, b0, c0);
        c1 = wmma_f32(a, b1, c1);
        c2 = wmma_f32(a, b2, c2);
        c3 = wmma_f32(a, b3, c3);
        a = an; b0 = bn0; b1 = bn1; b2 = bn2; b3 = bn3;
    }
    c0 = wmma_f32(a, b0, c0);
    c1 = wmma_f32(a, b1, c1);
    c2 = wmma_f32(a, b2, c2);
    c3 = wmma_f32(a, b3, c3);

    float* yout = Y + (size_t)(m0 + 8 * half) * N + n0 + l16;
#pragma unroll
    for (int r = 0; r < 8; ++r) {
        yout[(size_t)r * N +  0] = c0[r];
        yout[(size_t)r * N + 16] = c1[r];
        yout[(size_t)r * N + 32] = c2[r];
        yout[(size_t)r * N + 48] = c3[r];
    }
}

// ---------------------------------------------------------------------------
// Fused attention row-block kernel. One block = (b, h, 16 query rows).
//   Phase 1: scores 16x2048 = QK^T/8 + bias -> LDS (WMMA, A-frag reused 16x)
//   Phase 2: softmax in LDS (float4, wave32 shfl), single b128 HBM attn write
//   Phase 3: O = P @ V from LDS, split-K over wave halves, LDS combine
// ---------------------------------------------------------------------------
__global__ void attn_fused_kernel(const float* __restrict__ Q,    // [B*S, D]
                                  const float* __restrict__ Km,   // [B*S, D]
                                  const float* __restrict__ V,    // [B*S, D]
                                  const float* __restrict__ tab,  // [H, TABW]
                                  float* __restrict__ attn,       // [B,H,S,S]
                                  float* __restrict__ O) {        // [B*S, D]
    extern __shared__ float lds[];
    float* scratch = lds + 16 * LDSTRIDE;

    int bid = blockIdx.x;
    int qt = bid & 127; bid >>= 7;
    int h = bid % H_;
    int b = bid / H_;
    int m0 = qt << 4;

    int wave = threadIdx.x >> 5;
    int lane = threadIdx.x & 31;
    int half = lane >> 4, l16 = lane & 15;

    // ---------------- Phase 1: score tiles into LDS ----------------
    const float* arow = Q + (size_t)(b * S_ + m0 + l16) * D_ + h * KS_ + 2 * half;
    v2f afrag[16];
#pragma unroll
    for (int i = 0; i < 16; ++i) afrag[i] = *(const v2f*)(arow + 4 * i);

    const float* btab = tab + (size_t)h * TABW_;

    for (int j = 0; j < 16; ++j) {
        int n0 = (wave * 16 + j) << 4;
        const float* brow = Km + (size_t)(b * S_ + n0 + l16) * D_ + h * KS_ + 2 * half;
        v8f c = {};
        v2f bb = *(const v2f*)(brow);
#pragma unroll
        for (int i = 0; i < 15; ++i) {      // prefetch next frag, compute current
            v2f bn = *(const v2f*)(brow + 4 * (i + 1));
            c = wmma_f32(afrag[i], bb, c);
            bb = bn;
        }
        c = wmma_f32(afrag[15], bb, c);

        int n = n0 + l16;
#pragma unroll
        for (int r = 0; r < 8; ++r) {
            int m = m0 + r + 8 * half;
            int rel = n - m;
            rel = rel < -MAXD_ ? -MAXD_ : (rel > MAXD_ ? MAXD_ : rel);
            lds[(r + 8 * half) * LDSTRIDE + n] = c[r] * 0.125f + btab[rel + MAXD_];
        }
    }
    __syncthreads();

    // ---------------- Phase 2: softmax (2 rows per wave, float4) -----------
    float* attnbase = attn + ((size_t)(b * H_ + h) * S_ + m0) * S_;
#pragma unroll
    for (int rr = 0; rr < 2; ++rr) {
        int row = wave * 2 + rr;
        float* prow = lds + row * LDSTRIDE;

        float mx = -3.0e38f;
#pragma unroll
        for (int jj = 0; jj < 16; ++jj) {
            v4f t = *(const v4f*)(prow + 4 * lane + 128 * jj);
            mx = fmaxf(mx, fmaxf(fmaxf(t.x, t.y), fmaxf(t.z, t.w)));
        }
#pragma unroll
        for (int o = 16; o > 0; o >>= 1) mx = fmaxf(mx, __shfl_xor(mx, o, 32));

        float sum = 0.f;
#pragma unroll
        for (int jj = 0; jj < 16; ++jj) {
            v4f t = *(const v4f*)(prow + 4 * lane + 128 * jj);
            t.x = __expf(t.x - mx);
            t.y = __expf(t.y - mx);
            t.z = __expf(t.z - mx);
            t.w = __expf(t.w - mx);
            *(v4f*)(prow + 4 * lane + 128 * jj) = t;
            sum += (t.x + t.y) + (t.z + t.w);
        }
#pragma unroll
        for (int o = 16; o > 0; o >>= 1) sum += __shfl_xor(sum, o, 32);

        float inv = 1.f / sum;
        float* grow = attnbase + (size_t)row * S_;
#pragma unroll
        for (int jj = 0; jj < 16; ++jj) {
            v4f t = *(const v4f*)(prow + 4 * lane + 128 * jj);
            t.x *= inv; t.y *= inv; t.z *= inv; t.w *= inv;
            *(v4f*)(prow + 4 * lane + 128 * jj) = t;
            *(v4f*)(grow + 4 * lane + 128 * jj) = t;   // b128 coalesced HBM write
        }
    }
    __syncthreads();

    // ---------------- Phase 3: O = P @ V (split-K over wave halves) --------
    int nt = wave & 3;                 // 4 output tiles of 16 cols
    int kh = wave >> 2;                // K half: [0,1024) or [1024,2048)
    int n0 = nt << 4;

    const float* vbase = V + (size_t)(b * S_) * D_ + h * KS_ + n0 + l16;
    const float* abase = lds + l16 * LDSTRIDE + 2 * half;

    v8f c = {};
    int k0 = kh * (S_ / 2);

    // pipelined: prefetch next step's A (LDS) and V (global) fragments
    v2f a = *(const v2f*)(abase + k0);
    v2f bb;
    bb.x = vbase[(size_t)(k0 + 2 * half)     * D_];
    bb.y = vbase[(size_t)(k0 + 2 * half + 1) * D_];

    for (int kk = k0 + 4; kk < k0 + S_ / 2; kk += 4) {
        v2f an = *(const v2f*)(abase + kk);
        v2f bn;
        bn.x = vbase[(size_t)(kk + 2 * half)     * D_];
        bn.y = vbase[(size_t)(kk + 2 * half + 1) * D_];
        c = wmma_f32(a, bb, c);
        a = an; bb = bn;
    }
    c = wmma_f32(a, bb, c);

    if (kh == 1) {
#pragma unroll
        for (int r = 0; r < 8; ++r)
            scratch[nt * 256 + (r + 8 * half) * 16 + l16] = c[r];
    }
    __syncthreads();
    if (kh == 0) {
#pragma unroll
        for (int r = 0; r < 8; ++r)
            c[r] += scratch[nt * 256 + (r + 8 * half) * 16 + l16];
        float* yout = O + (size_t)(b * S_ + m0 + 8 * half) * D_ + h * KS_ + n0 + l16;
#pragma unroll
        for (int r = 0; r < 8; ++r) yout[(size_t)r * D_] = c[r];
    }
}

// ---------------------------------------------------------------------------
extern "C" void kernel_launch(void* const* d_in, const int* in_sizes, int n_in,
                              void* d_out, int out_size, void* d_ws, size_t ws_size,
                              hipStream_t stream) {
    (void)in_sizes; (void)n_in; (void)out_size; (void)ws_size;

    const float* x    = (const float*)d_in[0];  // [B,S,D]
    const float* Wq   = (const float*)d_in[1];  // [H*KS, D]
    const float* Wk   = (const float*)d_in[2];
    const float* Wv   = (const float*)d_in[3];
    const float* Wo   = (const float*)d_in[4];  // [D, H*VS]
    const float* bwts = (const float*)d_in[5];  // [H, NB]
    const float* bfea = (const float*)d_in[6];  // [TABW, NB]

    float* out  = (float*)d_out;                      // [B,S,D]
    float* attn = out + (size_t)B_ * S_ * D_;         // [B,H,S,S]

    const size_t QKV = (size_t)B_ * S_ * D_;          // 3,145,728 floats
    float* q   = (float*)d_ws;
    float* k   = q  + QKV;
    float* v   = k  + QKV;
    float* o1  = v  + QKV;
    float* tab = o1 + QKV;                            // [H, TABW]

    const int M = B_ * S_;                            // 4096

    // 1) relative-position bias table
    bias_kernel<<<(H_ * TABW_ + 255) / 256, 256, 0, stream>>>(bwts, bfea, tab);

    // 2) Q/K/V projections: 16x64 strip per wave -> 3072 waves / 768 blocks
    const int gemm_blocks = (M / 16) * (D_ / 64) / 4;
    gemm_xwT4_kernel<<<gemm_blocks, 128, 0, stream>>>(x, Wq, q, M, D_, D_);
    gemm_xwT4_kernel<<<gemm_blocks, 128, 0, stream>>>(x, Wk, k, M, D_, D_);
    gemm_xwT4_kernel<<<gemm_blocks, 128, 0, stream>>>(x, Wv, v, M, D_, D_);

    // 3) fused scores + bias + softmax + P@V  (one block per 16 query rows)
    const size_t lds_bytes = (size_t)LDS_FLOATS * sizeof(float);  // ~132.3 KB
    attn_fused_kernel<<<B_ * H_ * (S_ / 16), 256, lds_bytes, stream>>>(
        q, k, v, tab, attn, o1);

    // 4) output projection
    gemm_xwT4_kernel<<<gemm_blocks, 128, 0, stream>>>(o1, Wo, out, M, D_, D_);
}